// ImprovedGNN_79216376808020
// MI455X (gfx1250) — compile-verified
//
#include <hip/hip_runtime.h>

// ---------------------------------------------------------------------------
// ImprovedGNN for MI455X (gfx1250, wave32).
//   - FP32 WMMA (v_wmma_f32_16x16x4_f32) for all dense GEMMs
//   - weight stripes staged global->LDS column-major via async-to-LDS
//     (ASYNCcnt path) so every B fragment is a single aligned ds_load_b64
//   - edge pipeline uses b128 gathers; head-mean folded into scatter
// ---------------------------------------------------------------------------

typedef float v2f __attribute__((ext_vector_type(2)));
typedef float v8f __attribute__((ext_vector_type(8)));

#define DEVFN __device__ __forceinline__

constexpr int N_NODES = 40000;
constexpr int N_EDGES = 320000;
constexpr int ET      = N_EDGES + N_NODES;   // edges incl. self loops
constexpr int IN_DIM  = 256;
constexpr int HID     = 128;
constexpr int HEADS   = 4;
constexpr int N_GAT   = 3;
constexpr float EPS       = 1e-5f;
constexpr float NEG_SLOPE = 0.2f;

// ---------------- helpers ----------------

DEVFN unsigned f32_ord(float f) {              // monotone float->uint (for atomicMax)
    unsigned u = __float_as_uint(f);
    return (u & 0x80000000u) ? ~u : (u | 0x80000000u);
}
DEVFN float ord_f32(unsigned u) {
    return (u & 0x80000000u) ? __uint_as_float(u & 0x7FFFFFFFu) : __uint_as_float(~u);
}
// ord(-inf) = 0x007FFFFF -> init value for the running max

DEVFN void edge_sd(const int* __restrict__ ei, int j, int& s, int& d) {
    if (j < N_EDGES) { s = ei[j]; d = ei[N_EDGES + j]; }
    else             { s = j - N_EDGES; d = s; }       // self loop
}

DEVFN float wave_sum(float v) {
    #pragma unroll
    for (int off = 16; off > 0; off >>= 1) v += __shfl_xor(v, off, 32);
    return v;
}

// async global->LDS copies (gfx1250 ASYNCcnt path)
DEVFN void async_copy_b32(unsigned lds_off, const void* gptr) {
    asm volatile("global_load_async_to_lds_b32 %0, %1, off"
                 :: "v"(lds_off), "v"((unsigned long long)(size_t)gptr)
                 : "memory");
}
DEVFN void wait_asynccnt0() {
    asm volatile("s_wait_asynccnt 0" ::: "memory");
}

// ---------------- fills ----------------

__global__ void fill_f32(float* p, float v, int n) {
    int i = blockIdx.x * blockDim.x + threadIdx.x;
    if (i < n) p[i] = v;
}
__global__ void fill_u32(unsigned* p, unsigned v, int n) {
    int i = blockIdx.x * blockDim.x + threadIdx.x;
    if (i < n) p[i] = v;
}

// ---------------- FP32 WMMA GEMM:  C[M,N] = A[M,K] @ W[K,N] (+ bias) --------
// Block = 8 waves sharing one 64-column stripe of W. The stripe is staged
// COLUMN-MAJOR in LDS (padded K+2 rows per column -> bank-conflict-free) via
// async-to-LDS b32 scatter, so each B fragment {k,k+1} of a column is one
// aligned ds_load_b64 feeding WMMA directly (no register shuffles).
// Fragment layouts per CDNA5 ISA 7.12.2 (16x4 f32 A, 4x16 f32 B, 16x16 f32 D).

template <int K, int N>
__global__ __launch_bounds__(256)
void wmma_gemm_f32(const float* __restrict__ A,
                   const float* __restrict__ W,
                   const float* __restrict__ bias,
                   float* __restrict__ C, int M) {
    constexpr int stripsPerRow = N / 64;
    constexpr int KP = K + 2;                       // padded column height
    const int ns = blockIdx.x % stripsPerRow;       // which 64-col stripe
    const int mg = blockIdx.x / stripsPerRow;       // which group of 8 m-strips
    const int n0 = ns * 64;
    const int waveId = threadIdx.x >> 5;
    const int lane   = threadIdx.x & 31;
    const int m0 = (mg * 8 + waveId) * 16;

    // ---- stage W[:, n0:n0+64] into LDS column-major via async-to-LDS ----
    __shared__ float Wl[64 * KP];
    {
        const unsigned lds_base = (unsigned)(size_t)&Wl[0]; // low 32 bits = LDS offset
        const float* Wg = W + n0;
        // K*64 elements; consecutive lanes take consecutive columns of a row
        // -> coalesced global reads, conflict-free LDS scatter (stride KP).
        for (int c = threadIdx.x; c < K * 64; c += 256) {
            int row = c >> 6;
            int col = c & 63;
            async_copy_b32(lds_base + (unsigned)((col * KP + row) * 4),
                           Wg + (size_t)row * N + col);
        }
        wait_asynccnt0();
        __syncthreads();
    }

    if (m0 < M) {
        const int cl    = lane & 15;
        const int khalf = (lane >> 4) * 2;  // lanes 0-15: K={k,k+1}; 16-31: {k+2,k+3}
        const float* Arow = A + (size_t)(m0 + cl) * K + khalf;
        const float* B0 = &Wl[(cl     ) * KP + khalf];
        const float* B1 = &Wl[(cl + 16) * KP + khalf];
        const float* B2 = &Wl[(cl + 32) * KP + khalf];
        const float* B3 = &Wl[(cl + 48) * KP + khalf];

        v8f acc0 = {}, acc1 = {}, acc2 = {}, acc3 = {};

        for (int k = 0; k < K; k += 4) {
            v2f a  = *(const v2f*)(Arow + k);   // global b64
            v2f b0 = *(const v2f*)(B0 + k);     // LDS ds_load_b64, aligned pairs
            v2f b1 = *(const v2f*)(B1 + k);
            v2f b2 = *(const v2f*)(B2 + k);
            v2f b3 = *(const v2f*)(B3 + k);
            acc0 = __builtin_amdgcn_wmma_f32_16x16x4_f32(false, a, false, b0, (short)0, acc0, false, false);
            acc1 = __builtin_amdgcn_wmma_f32_16x16x4_f32(false, a, false, b1, (short)0, acc1, false, false);
            acc2 = __builtin_amdgcn_wmma_f32_16x16x4_f32(false, a, false, b2, (short)0, acc2, false, false);
            acc3 = __builtin_amdgcn_wmma_f32_16x16x4_f32(false, a, false, b3, (short)0, acc3, false, false);
        }

        float bb0 = 0.f, bb1 = 0.f, bb2 = 0.f, bb3 = 0.f;
        if (bias) {
            bb0 = bias[n0 + cl];      bb1 = bias[n0 + 16 + cl];
            bb2 = bias[n0 + 32 + cl]; bb3 = bias[n0 + 48 + cl];
        }
        const int rbase = m0 + ((lane >> 4) << 3);  // D: lanes16-31 rows m0+8..15
        #pragma unroll
        for (int r = 0; r < 8; ++r) {
            size_t base = (size_t)(rbase + r) * N + n0 + cl;
            C[base]      = acc0[r] + bb0;
            C[base + 16] = acc1[r] + bb1;
            C[base + 32] = acc2[r] + bb2;
            C[base + 48] = acc3[r] + bb3;
        }
    }
}

// ---------------- attention coefficients: a_s/a_d [N, HEADS] ----------------
// wave per (node, head); one float4 per lane covers all 128 channels.

__global__ void att_coef_kernel(const float* __restrict__ xh,
                                const float* __restrict__ att_src,
                                const float* __restrict__ att_dst,
                                float* __restrict__ a_s, float* __restrict__ a_d) {
    int task = blockIdx.x * (blockDim.x >> 5) + (threadIdx.x >> 5);  // node*4+head
    int lane = threadIdx.x & 31;
    if (task >= N_NODES * HEADS) return;
    int n = task >> 2, hd = task & 3;
    const float* xr = xh + (size_t)n * (HEADS * HID) + hd * HID;
    float4 v  = *(const float4*)(xr + (lane << 2));
    float4 va = *(const float4*)(att_src + hd * HID + (lane << 2));
    float4 vd = *(const float4*)(att_dst + hd * HID + (lane << 2));
    float ss = v.x * va.x + v.y * va.y + v.z * va.z + v.w * va.w;
    float sd = v.x * vd.x + v.y * vd.y + v.z * vd.z + v.w * vd.w;
    ss = wave_sum(ss);
    sd = wave_sum(sd);
    if (lane == 0) { a_s[task] = ss; a_d[task] = sd; }
}

// ---------------- segment softmax pass 1: e + running max -------------------

__global__ void edge_max_kernel(const int* __restrict__ ei,
                                const float* __restrict__ a_s,
                                const float* __restrict__ a_d,
                                float* __restrict__ ebuf,
                                unsigned* __restrict__ m_u) {
    int tid = blockIdx.x * blockDim.x + threadIdx.x;   // == j*4 + h
    if (tid >= ET * HEADS) return;
    int j = tid >> 2, hd = tid & 3;
    int s, d; edge_sd(ei, j, s, d);
    float e = a_s[s * HEADS + hd] + a_d[d * HEADS + hd];
    e = (e > 0.f) ? e : NEG_SLOPE * e;                 // leaky relu
    ebuf[tid] = e;
    atomicMax(&m_u[d * HEADS + hd], f32_ord(e));
}

// ---------------- pass 2: exp(e - max), denom -------------------------------

__global__ void edge_sum_kernel(const int* __restrict__ ei,
                                float* __restrict__ ebuf,
                                const unsigned* __restrict__ m_u,
                                float* __restrict__ denom) {
    int tid = blockIdx.x * blockDim.x + threadIdx.x;
    if (tid >= ET * HEADS) return;
    int j = tid >> 2, hd = tid & 3;
    int s, d; edge_sd(ei, j, s, d);
    float mm = ord_f32(m_u[d * HEADS + hd]);
    float ex = expf(ebuf[tid] - mm);
    ebuf[tid] = ex;
    atomicAdd(&denom[d * HEADS + hd], ex);
}

// ---------------- pass 3: weighted aggregation (head-mean folded in) --------
// wave per edge; lane owns 4 consecutive channels -> b128 gathers.

__global__ void edge_agg_kernel(const int* __restrict__ ei,
                                const float* __restrict__ xh,
                                const float* __restrict__ ebuf,
                                const float* __restrict__ denom,
                                float* __restrict__ acc) {
    int j = blockIdx.x * (blockDim.x >> 5) + (threadIdx.x >> 5);
    int lane = threadIdx.x & 31;
    if (j >= ET) return;
    int s, d; edge_sd(ei, j, s, d);
    float al = 0.f;
    if (lane < HEADS)
        al = ebuf[j * HEADS + lane] / denom[d * HEADS + lane] * 0.25f;  // /HEADS
    float a0 = __shfl(al, 0, 32), a1 = __shfl(al, 1, 32);
    float a2 = __shfl(al, 2, 32), a3 = __shfl(al, 3, 32);
    const float* xs = xh + (size_t)s * (HEADS * HID);
    const int c4 = lane << 2;                        // 4 consecutive channels
    float4 v0 = *(const float4*)(xs + 0 * HID + c4); // b128 gathers
    float4 v1 = *(const float4*)(xs + 1 * HID + c4);
    float4 v2 = *(const float4*)(xs + 2 * HID + c4);
    float4 v3 = *(const float4*)(xs + 3 * HID + c4);
    float r0 = a0 * v0.x + a1 * v1.x + a2 * v2.x + a3 * v3.x;
    float r1 = a0 * v0.y + a1 * v1.y + a2 * v2.y + a3 * v3.y;
    float r2 = a0 * v0.z + a1 * v1.z + a2 * v2.z + a3 * v3.z;
    float r3 = a0 * v0.w + a1 * v1.w + a2 * v2.w + a3 * v3.w;
    float* ad = acc + (size_t)d * HID + c4;
    atomicAdd(&ad[0], r0);
    atomicAdd(&ad[1], r1);
    atomicAdd(&ad[2], r2);
    atomicAdd(&ad[3], r3);
}

// ---------------- per-node epilogue: bias + LayerNorm + ReLU + residual -----

__global__ void epilogue_kernel(const float* __restrict__ acc,
                                const float* __restrict__ bias,
                                const float* __restrict__ gamma,
                                const float* __restrict__ beta,
                                float* __restrict__ h) {
    int n = blockIdx.x;
    int c = threadIdx.x;                         // 128 threads = 4 waves
    float v = acc[(size_t)n * HID + c] + bias[c];
    float s = wave_sum(v);
    float s2 = wave_sum(v * v);
    __shared__ float ls[4], ls2[4];
    int w = threadIdx.x >> 5, lane = threadIdx.x & 31;
    if (lane == 0) { ls[w] = s; ls2[w] = s2; }
    __syncthreads();
    float S  = ls[0] + ls[1] + ls[2] + ls[3];
    float S2 = ls2[0] + ls2[1] + ls2[2] + ls2[3];
    float mu  = S * (1.f / HID);
    float var = S2 * (1.f / HID) - mu * mu;
    float hn = (v - mu) * rsqrtf(var + EPS) * gamma[c] + beta[c];
    hn = fmaxf(hn, 0.f);
    h[(size_t)n * HID + c] += hn;                // + identity (h still holds it)
}

// ---------------- GCN head --------------------------------------------------

__global__ void deg_kernel(const int* __restrict__ ei, float* __restrict__ deg) {
    int j = blockIdx.x * blockDim.x + threadIdx.x;
    if (j >= ET) return;
    int s, d; edge_sd(ei, j, s, d);
    atomicAdd(&deg[d], 1.0f);
}

__global__ void xw_kernel(const float* __restrict__ h,
                          const float* __restrict__ gcn_W,
                          float* __restrict__ xw) {
    int n = blockIdx.x * (blockDim.x >> 5) + (threadIdx.x >> 5);
    int lane = threadIdx.x & 31;
    if (n >= N_NODES) return;
    float4 v = *(const float4*)(h + (size_t)n * HID + (lane << 2));
    float4 w = *(const float4*)(gcn_W + (lane << 2));
    float sum = v.x * w.x + v.y * w.y + v.z * w.z + v.w * w.w;
    sum = wave_sum(sum);
    if (lane == 0) xw[n] = sum;
}

__global__ void gcn_edge_kernel(const int* __restrict__ ei,
                                const float* __restrict__ xw,
                                const float* __restrict__ deg,
                                float* __restrict__ out) {
    int j = blockIdx.x * blockDim.x + threadIdx.x;
    if (j >= ET) return;
    int s, d; edge_sd(ei, j, s, d);
    float norm = rsqrtf(deg[s]) * rsqrtf(deg[d]);
    atomicAdd(&out[d], xw[s] * norm);
}

__global__ void sigmoid_kernel(float* __restrict__ out,
                               const float* __restrict__ gcn_b, int n) {
    int i = blockIdx.x * blockDim.x + threadIdx.x;
    if (i >= n) return;
    float v = out[i] + gcn_b[0];
    out[i] = 1.f / (1.f + expf(-v));
}

// ---------------- launch ----------------------------------------------------

extern "C" void kernel_launch(void* const* d_in, const int* in_sizes, int n_in,
                              void* d_out, int out_size, void* d_ws, size_t ws_size,
                              hipStream_t stream) {
    const float* x        = (const float*)d_in[0];
    const int*   ei       = (const int*)d_in[1];     // int32 (jax default x64-off)
    const float* proj_W   = (const float*)d_in[2];
    const float* proj_b   = (const float*)d_in[3];
    const float* gat_W    = (const float*)d_in[4];
    const float* att_src  = (const float*)d_in[5];
    const float* att_dst  = (const float*)d_in[6];
    const float* gat_bias = (const float*)d_in[7];
    const float* ln_gamma = (const float*)d_in[8];
    const float* ln_beta  = (const float*)d_in[9];
    const float* gcn_W    = (const float*)d_in[10];
    const float* gcn_b    = (const float*)d_in[11];
    float* out = (float*)d_out;

    // workspace layout (~131 MB)
    float*    h     = (float*)d_ws;                       // [N, HID]
    float*    xh    = h + (size_t)N_NODES * HID;          // [N, HEADS*HID]
    float*    a_s   = xh + (size_t)N_NODES * HEADS * HID; // [N, HEADS]
    float*    a_d   = a_s + N_NODES * HEADS;              // [N, HEADS]
    unsigned* m_u   = (unsigned*)(a_d + N_NODES * HEADS); // [N, HEADS]
    float*    denom = (float*)(m_u + N_NODES * HEADS);    // [N, HEADS]
    float*    ebuf  = denom + N_NODES * HEADS;            // [ET, HEADS]
    float*    acc   = ebuf + (size_t)ET * HEADS;          // [N, HID]
    float*    deg   = acc + (size_t)N_NODES * HID;        // [N]
    float*    xw    = deg + N_NODES;                      // [N]

    const int mStrips = N_NODES / 16;                     // 2500
    const int mGroups = (mStrips + 7) / 8;                // 313 (8 waves / block)

    // 1) projection: h = x @ proj_W + proj_b
    wmma_gemm_f32<IN_DIM, HID><<<mGroups * (HID / 64), 256, 0, stream>>>(
        x, proj_W, proj_b, h, N_NODES);

    // 2) GAT layers
    for (int i = 0; i < N_GAT; ++i) {
        const float* Wi = gat_W + (size_t)i * HID * HEADS * HID;
        wmma_gemm_f32<HID, HEADS * HID><<<mGroups * ((HEADS * HID) / 64), 256, 0, stream>>>(
            h, Wi, nullptr, xh, N_NODES);
        att_coef_kernel<<<(N_NODES * HEADS + 7) / 8, 256, 0, stream>>>(
            xh, att_src + i * HEADS * HID, att_dst + i * HEADS * HID, a_s, a_d);
        fill_u32<<<(N_NODES * HEADS + 255) / 256, 256, 0, stream>>>(
            m_u, 0x007FFFFFu /* ord(-inf) */, N_NODES * HEADS);
        fill_f32<<<(N_NODES * HEADS + 255) / 256, 256, 0, stream>>>(
            denom, 0.f, N_NODES * HEADS);
        fill_f32<<<(N_NODES * HID + 255) / 256, 256, 0, stream>>>(
            acc, 0.f, N_NODES * HID);
        edge_max_kernel<<<(ET * HEADS + 255) / 256, 256, 0, stream>>>(
            ei, a_s, a_d, ebuf, m_u);
        edge_sum_kernel<<<(ET * HEADS + 255) / 256, 256, 0, stream>>>(
            ei, ebuf, m_u, denom);
        edge_agg_kernel<<<(ET + 7) / 8, 256, 0, stream>>>(
            ei, xh, ebuf, denom, acc);
        epilogue_kernel<<<N_NODES, HID, 0, stream>>>(
            acc, gat_bias + i * HID, ln_gamma + i * HID, ln_beta + i * HID, h);
    }

    // 3) GCN head
    fill_f32<<<(N_NODES + 255) / 256, 256, 0, stream>>>(deg, 0.f, N_NODES);
    fill_f32<<<(N_NODES + 255) / 256, 256, 0, stream>>>(out, 0.f, N_NODES);
    deg_kernel<<<(ET + 255) / 256, 256, 0, stream>>>(ei, deg);
    xw_kernel<<<(N_NODES + 7) / 8, 256, 0, stream>>>(h, gcn_W, xw);
    gcn_edge_kernel<<<(ET + 255) / 256, 256, 0, stream>>>(ei, xw, deg, out);
    sigmoid_kernel<<<(N_NODES + 255) / 256, 256, 0, stream>>>(out, gcn_b, N_NODES);
}